// GCNDeepIndepNormal_9689446220632
// MI455X (gfx1250) — compile-verified
//
#include <hip/hip_runtime.h>
#include <hip/hip_bf16.h>

// ---------------------------------------------------------------------------
// GCN layer: h = relu(x@Wb+bb); agg = D^-1/2 (A+I) D^-1/2 h;
//            mean = agg@Wm+bm; std = exp(0.5*(agg@Wl+bl))
// fp32 throughout; GEMMs on the CDNA5 matrix pipe via V_WMMA_F32_16X16X4_F32.
// ---------------------------------------------------------------------------

#define N_NODES 100000
#define IN_FEAT 64
#define HIDDEN  64
#define OUTF    32

typedef __attribute__((ext_vector_type(2))) float v2f;
typedef __attribute__((ext_vector_type(8))) float v8f;

// ----------------------- degree / normalization ---------------------------

__global__ void deg_init_kernel(float* __restrict__ deg) {
    int i = blockIdx.x * blockDim.x + threadIdx.x;
    if (i < N_NODES) deg[i] = 1.0f;              // self-loop contribution
}

__global__ void deg_accum_kernel(const int* __restrict__ tgt,
                                 float* __restrict__ deg, int E) {
    int e = blockIdx.x * blockDim.x + threadIdx.x;
    if (e < E) atomicAdd(&deg[tgt[e]], 1.0f);    // degree over TARGETS only
}

__global__ void deg_to_dinv_kernel(float* __restrict__ deg) {
    int i = blockIdx.x * blockDim.x + threadIdx.x;
    if (i < N_NODES) deg[i] = rsqrtf(deg[i]);
}

// ----------------------- backbone GEMM (WMMA f32) -------------------------
// One wave computes a 16x16 tile of h = relu(x @ Wb + bb).
// Block = 256 threads = 8 waves = 2 row-tiles x 4 col-tiles = 32 rows x 64 cols.
// N_NODES = 100000 = 3125 * 32 exactly -> no tail handling.
//
// A-frag (16x4 f32, ISA 7.12.2): lanes 0-15 hold row M=lane, K=k0..k0+1;
//                                lanes 16-31 hold row M=lane-16, K=k0+2..k0+3.
// B-frag (4x16 f32): half-wave hi=0 holds rows K=k0,k0+1; hi=1 holds K=k0+2,k0+3;
//                    column N = lane&15 within the VGPR.
// C/D (16x16 f32): vgpr v, lanes 0-15 -> M=v, N=lane; lanes 16-31 -> M=v+8.

__global__ __launch_bounds__(256) void backbone_wmma_kernel(
    const float* __restrict__ x, const float* __restrict__ Wb,
    const float* __restrict__ bb, float* __restrict__ h) {
    const int wave = threadIdx.x >> 5;
    const int lane = threadIdx.x & 31;
    const int rt   = blockIdx.x * 2 + (wave >> 2);   // 16-row tile index
    const int ct   = wave & 3;                       // 16-col tile index (of 4)
    const int row0 = rt * 16;
    const int col0 = ct * 16;
    const int l15  = lane & 15;
    const int hi   = lane >> 4;                      // which half-wave (K split)

    v8f acc = {};
    #pragma unroll
    for (int kb = 0; kb < IN_FEAT; kb += 4) {
        const int k = kb + 2 * hi;
        // A fragment: two consecutive K values of this lane's row
        const float* ap = x + (size_t)(row0 + l15) * IN_FEAT + k;
        v2f a; a.x = ap[0]; a.y = ap[1];
        // B fragment: Wb rows k, k+1 at column col0+l15
        v2f b;
        b.x = Wb[(size_t)(k + 0) * HIDDEN + col0 + l15];
        b.y = Wb[(size_t)(k + 1) * HIDDEN + col0 + l15];
        acc = __builtin_amdgcn_wmma_f32_16x16x4_f32(
            false, a, false, b, (short)0, acc, false, false);
    }

    const float bias = bb[col0 + l15];
    #pragma unroll
    for (int v = 0; v < 8; ++v) {
        const int r = row0 + v + 8 * hi;
        float val = acc[v] + bias;
        val = val > 0.0f ? val : 0.0f;               // relu
        h[(size_t)r * HIDDEN + col0 + l15] = val;
    }
}

// ----------------------- aggregation --------------------------------------

// Self-loop term: agg[i] = h[i] * dinv[i]^2  (also initializes agg buffer)
__global__ void agg_self_kernel(const float* __restrict__ h,
                                const float* __restrict__ dinv,
                                float* __restrict__ agg) {
    int tid = blockIdx.x * blockDim.x + threadIdx.x;   // over N_NODES*64
    int node = tid >> 6;
    float d = dinv[node];
    agg[tid] = h[tid] * d * d;
}

// Edge scatter-add: 64 threads per edge, one feature each. agg/h are 25.6MB
// each -> random accesses stay resident in the 192MB L2.
__global__ void edge_scatter_kernel(const int* __restrict__ src,
                                    const int* __restrict__ tgt,
                                    const float* __restrict__ h,
                                    const float* __restrict__ dinv,
                                    float* __restrict__ agg, int E) {
    long long tid = (long long)blockIdx.x * blockDim.x + threadIdx.x;
    long long e = tid >> 6;
    if (e >= E) return;
    int f = (int)(tid & 63);
    int s = src[e];
    int t = tgt[e];
    float norm = dinv[s] * dinv[t];
    atomicAdd(&agg[(size_t)t * HIDDEN + f], h[(size_t)s * HIDDEN + f] * norm);
}

// ----------------------- head GEMMs (WMMA f32) ----------------------------
// Block = 8 waves: 2 row-tiles x 2 col-tiles x 2 heads (mean / logvar).
// out layout: [mean (100000x32) | std (100000x32)] concatenated flat.

__global__ __launch_bounds__(256) void heads_wmma_kernel(
    const float* __restrict__ agg,
    const float* __restrict__ Wm, const float* __restrict__ bm,
    const float* __restrict__ Wl, const float* __restrict__ bl,
    float* __restrict__ out) {
    const int wave = threadIdx.x >> 5;
    const int lane = threadIdx.x & 31;
    const int rt   = blockIdx.x * 2 + (wave >> 2);
    const int head = (wave >> 1) & 1;                // 0 = mean, 1 = logvar
    const int ct   = wave & 1;                       // 16-col tile (of 2)
    const float* __restrict__ W  = head ? Wl : Wm;
    const float* __restrict__ bv = head ? bl : bm;
    const int row0 = rt * 16;
    const int col0 = ct * 16;
    const int l15  = lane & 15;
    const int hi   = lane >> 4;

    v8f acc = {};
    #pragma unroll
    for (int kb = 0; kb < HIDDEN; kb += 4) {
        const int k = kb + 2 * hi;
        const float* ap = agg + (size_t)(row0 + l15) * HIDDEN + k;
        v2f a; a.x = ap[0]; a.y = ap[1];
        v2f b;
        b.x = W[(size_t)(k + 0) * OUTF + col0 + l15];
        b.y = W[(size_t)(k + 1) * OUTF + col0 + l15];
        acc = __builtin_amdgcn_wmma_f32_16x16x4_f32(
            false, a, false, b, (short)0, acc, false, false);
    }

    const float bias = bv[col0 + l15];
    const size_t head_off = (size_t)head * (size_t)N_NODES * OUTF;
    #pragma unroll
    for (int v = 0; v < 8; ++v) {
        const int r = row0 + v + 8 * hi;
        float val = acc[v] + bias;
        if (head) val = __expf(0.5f * val);          // std = exp(0.5*logvar)
        out[head_off + (size_t)r * OUTF + col0 + l15] = val;
    }
}

// ----------------------- launch -------------------------------------------

extern "C" void kernel_launch(void* const* d_in, const int* in_sizes, int n_in,
                              void* d_out, int out_size, void* d_ws, size_t ws_size,
                              hipStream_t stream) {
    const float* x  = (const float*)d_in[0];
    const int*   ei = (const int*)d_in[1];   // edge_index, [2, E] flat
    const float* Wb = (const float*)d_in[2];
    const float* bb = (const float*)d_in[3];
    const float* Wm = (const float*)d_in[4];
    const float* bm = (const float*)d_in[5];
    const float* Wl = (const float*)d_in[6];
    const float* bl = (const float*)d_in[7];
    float* out = (float*)d_out;

    const int E = in_sizes[1] / 2;
    const int* src = ei;         // edge_index[0]
    const int* tgt = ei + E;     // edge_index[1]

    // Workspace layout (fp32 elements):
    //   dinv : [0, 102400)            (N_NODES rounded up, used as deg->dinv)
    //   h    : [102400, +N*64)
    //   agg  : [102400+N*64, +N*64)   total ~51.6 MB
    float* ws   = (float*)d_ws;
    float* dinv = ws;
    float* h    = ws + 102400;
    float* agg  = h + (size_t)N_NODES * HIDDEN;

    // 1) degrees (targets + self loops) -> dinv = rsqrt(deg)
    deg_init_kernel<<<(N_NODES + 255) / 256, 256, 0, stream>>>(dinv);
    deg_accum_kernel<<<(E + 255) / 256, 256, 0, stream>>>(tgt, dinv, E);
    deg_to_dinv_kernel<<<(N_NODES + 255) / 256, 256, 0, stream>>>(dinv);

    // 2) backbone GEMM + relu (WMMA f32): 32 rows per block
    backbone_wmma_kernel<<<N_NODES / 32, 256, 0, stream>>>(x, Wb, bb, h);

    // 3) aggregation: self-loop init, then edge scatter-add (L2-resident)
    agg_self_kernel<<<(N_NODES * HIDDEN) / 256, 256, 0, stream>>>(h, dinv, agg);
    long long ethreads = (long long)E * 64;
    int eblocks = (int)((ethreads + 255) / 256);
    edge_scatter_kernel<<<eblocks, 256, 0, stream>>>(src, tgt, h, dinv, agg, E);

    // 4) both heads fused (WMMA f32) + bias + exp for std
    heads_wmma_kernel<<<N_NODES / 32, 256, 0, stream>>>(agg, Wm, bm, Wl, bl, out);
}